// GATModel_76836964926155
// MI455X (gfx1250) — compile-verified
//
#include <hip/hip_runtime.h>
#include <hip/hip_bf16.h>
#include <math.h>

typedef __attribute__((ext_vector_type(2))) float v2f;
typedef __attribute__((ext_vector_type(8))) float v8f;

#define NEG_SLOPE 0.2f

// ---------------------------------------------------------------- utilities

__global__ void k_fill(float* __restrict__ p, float v, long long n) {
    long long i = (long long)blockIdx.x * blockDim.x + threadIdx.x;
    if (i < n) p[i] = v;
}

// W[K][Nc] (row-major) -> Wt[Nc][K]  (i.e. column-major copy of W)
__global__ void k_transpose(const float* __restrict__ W, float* __restrict__ Wt,
                            int K, int Nc) {
    int t = blockIdx.x * blockDim.x + threadIdx.x;
    if (t >= K * Nc) return;
    const int k = t / Nc;
    const int c = t - k * Nc;
    Wt[c * K + k] = W[t];
}

// float atomic max via sign-magnitude monotonic int atomics.
// Requires destination initialized to -INF (0xFF800000).
__device__ inline void atomicMaxF32(float* addr, float v) {
    if (v >= 0.0f)
        atomicMax((int*)addr, __float_as_int(v));
    else
        atomicMin((unsigned int*)addr, __float_as_uint(v));
}

// ------------------------------------------------------- fp32 WMMA GEMM
// C[M,NC] = A[M,128] @ W[128,NC], with W supplied TRANSPOSED (Wt[NC][128]).
// blockDim.x = 128 (4 waves); each wave computes one 16x16 tile.
// grid.x = ceil(M/16), grid.y = NC/64.
// VGPR layouts per CDNA5 ISA 7.12.2:
//   A 16x4  f32: lane L holds row L%16; elem0 = K + (L<16?0:2), elem1 = +1
//   B 4x16  f32: lane L holds col L%16; elem i = row (L<16?0:2)+i
//   C 16x16 f32: VGPR r, lane L -> row r + (L<16?0:8), col L%16
// With Wt, both A and B fragments are contiguous k-pairs -> b64 loads with
// compile-time immediate offsets (NC is a template parameter, K constexpr).
// Pad rows are CLAMPED to row M-1 on load and masked at store time ->
// no exec divergence in the K loop, EXEC stays all-1s (WMMA requirement).
template <int NC>
__global__ void k_gemm_wmma(const float* __restrict__ A,
                            const float* __restrict__ Wt,
                            float* __restrict__ C,
                            int M) {
    constexpr int K = 128;
    const int lane = threadIdx.x & 31;
    const int wave = threadIdx.x >> 5;
    const int l16  = lane & 15;
    const int hi   = lane >> 4;                 // 0: lanes 0-15, 1: lanes 16-31
    const int row0 = blockIdx.x * 16;
    const int col0 = blockIdx.y * 64 + wave * 16;
    int row = row0 + l16;
    if (row >= M) row = M - 1;                  // clamp (store is masked below)
    const int col = col0 + l16;

    const float* ap = A  + (size_t)row * K + (hi << 1);
    const float* bp = Wt + (size_t)col * K + (hi << 1);

    v8f acc0 = {}, acc1 = {};
#pragma unroll
    for (int kk = 0; kk < K; kk += 8) {
        v2f a0, a1, b0, b1;
        a0.x = ap[kk];     a0.y = ap[kk + 1];   // contiguous -> b64, imm offset
        a1.x = ap[kk + 4]; a1.y = ap[kk + 5];
        b0.x = bp[kk];     b0.y = bp[kk + 1];
        b1.x = bp[kk + 4]; b1.y = bp[kk + 5];
        acc0 = __builtin_amdgcn_wmma_f32_16x16x4_f32(
            false, a0, false, b0, (short)0, acc0, false, false);
        acc1 = __builtin_amdgcn_wmma_f32_16x16x4_f32(
            false, a1, false, b1, (short)0, acc1, false, false);
    }

#pragma unroll
    for (int r = 0; r < 8; ++r) {
        const int rr = row0 + r + hi * 8;
        if (rr < M) C[(size_t)rr * NC + col] = acc0[r] + acc1[r];
    }
}

// ---------------------------------------------- per-node attention logits
// as_out[n,h] = sum_c h[n,h,c]*a_s[h,c] ; ad_out likewise.
__global__ void k_alpha(const float* __restrict__ h,
                        const float* __restrict__ a_s,
                        const float* __restrict__ a_d,
                        float* __restrict__ as_out,
                        float* __restrict__ ad_out,
                        long long Nn, int H, int Cch) {
    long long t = (long long)blockIdx.x * blockDim.x + threadIdx.x;
    if (t >= Nn * H) return;
    const int hh = (int)(t % H);
    const long long n = t / H;
    const float* hp  = h + n * (long long)(H * Cch) + (long long)hh * Cch;
    const float* asp = a_s + (long long)hh * Cch;
    const float* adp = a_d + (long long)hh * Cch;
    float s = 0.0f, d = 0.0f;
    for (int c = 0; c < Cch; ++c) {
        const float v = hp[c];
        s += v * asp[c];
        d += v * adp[c];
    }
    as_out[t] = s;
    ad_out[t] = d;
}

// -------------------------------------------- edge pass 1: leaky + seg-max
__global__ void k_edge_max(const int* __restrict__ src,
                           const int* __restrict__ dst,
                           long long E, long long Nn, int H,
                           const float* __restrict__ as_,
                           const float* __restrict__ ad_,
                           float* __restrict__ ev,
                           float* __restrict__ m) {
    long long t = (long long)blockIdx.x * blockDim.x + threadIdx.x;
    const long long total = (E + Nn) * H;
    if (t >= total) return;
    const long long e = t / H;
    const int hh = (int)(t % H);
    long long s_, d_;
    if (e < E) { s_ = src[e]; d_ = dst[e]; } else { s_ = d_ = e - E; }
    float x = as_[s_ * H + hh] + ad_[d_ * H + hh];
    x = (x > 0.0f) ? x : NEG_SLOPE * x;
    ev[t] = x;
    atomicMaxF32(&m[d_ * H + hh], x);
}

// -------------------------------------- edge pass 2: exp(e - m) + seg-sum
__global__ void k_edge_exp(const int* __restrict__ src,
                           const int* __restrict__ dst,
                           long long E, long long Nn, int H,
                           const float* __restrict__ m,
                           float* __restrict__ ev,
                           float* __restrict__ den) {
    long long t = (long long)blockIdx.x * blockDim.x + threadIdx.x;
    const long long total = (E + Nn) * H;
    if (t >= total) return;
    const long long e = t / H;
    const int hh = (int)(t % H);
    long long d_;
    if (e < E) { d_ = dst[e]; } else { d_ = e - E; }
    const float x = expf(ev[t] - m[d_ * H + hh]);
    ev[t] = x;
    atomicAdd(&den[d_ * H + hh], x);
}

// ------------------------- edge pass 3: weighted gather + scatter-add
// one block per edge, one channel per thread (D = H*Cch threads).
__global__ void k_edge_scatter(const int* __restrict__ src,
                               const int* __restrict__ dst,
                               long long E, long long Nn, int H, int D,
                               const float* __restrict__ h,
                               const float* __restrict__ ev,
                               const float* __restrict__ den,
                               float* __restrict__ out) {
    const long long e = blockIdx.x;
    const int c = threadIdx.x;
    long long s_, d_;
    if (e < E) { s_ = src[e]; d_ = dst[e]; } else { s_ = d_ = e - E; }
    const int hh = c / (D / H);
    const float w = ev[e * H + hh] / den[d_ * H + hh];
    atomicAdd(&out[d_ * (long long)D + c], h[s_ * (long long)D + c] * w);
}

// ------------------------------------------------------- bias + ReLU
__global__ void k_bias_relu(float* __restrict__ x, const float* __restrict__ b,
                            long long Nn, int D) {
    long long t = (long long)blockIdx.x * blockDim.x + threadIdx.x;
    if (t >= Nn * D) return;
    const float v = x[t] + b[(int)(t % D)];
    x[t] = v > 0.0f ? v : 0.0f;
}

// -------------------- final: bias + ReLU + L2 normalize (D = 64 fixed)
// one wave32 per node; 2 channels per lane; shfl_xor tree reduction.
__global__ void k_final(const float* __restrict__ agg,
                        const float* __restrict__ b,
                        float* __restrict__ out, long long Nn) {
    const int lane = threadIdx.x & 31;
    const long long n = (long long)blockIdx.x * (blockDim.x >> 5) + (threadIdx.x >> 5);
    if (n >= Nn) return;
    float v0 = agg[n * 64 + lane] + b[lane];
    float v1 = agg[n * 64 + 32 + lane] + b[32 + lane];
    v0 = fmaxf(v0, 0.0f);
    v1 = fmaxf(v1, 0.0f);
    float ss = v0 * v0 + v1 * v1;
#pragma unroll
    for (int o = 16; o > 0; o >>= 1) ss += __shfl_xor(ss, o, 32);
    const float inv = 1.0f / fmaxf(sqrtf(ss), 1e-12f);
    out[n * 64 + lane] = v0 * inv;
    out[n * 64 + 32 + lane] = v1 * inv;
}

// --------------------------------------------------------------- launcher

static inline unsigned blocks_for(long long n, int bs) {
    return (unsigned)((n + bs - 1) / bs);
}

extern "C" void kernel_launch(void* const* d_in, const int* in_sizes, int n_in,
                              void* d_out, int out_size, void* d_ws, size_t ws_size,
                              hipStream_t stream) {
    const float* x   = (const float*)d_in[0];
    const int*   ei  = (const int*)d_in[1];
    const float* W1  = (const float*)d_in[3];
    const float* as1 = (const float*)d_in[4];
    const float* ad1 = (const float*)d_in[5];
    const float* b1  = (const float*)d_in[6];
    const float* W2  = (const float*)d_in[7];
    const float* as2 = (const float*)d_in[8];
    const float* ad2 = (const float*)d_in[9];
    const float* b2  = (const float*)d_in[10];
    float* out = (float*)d_out;

    const long long Nn = in_sizes[0] / 128;   // 100000
    const long long Ee = in_sizes[1] / 2;     // 1600000
    const long long TE = Ee + Nn;             // edges + self loops
    const int* srcp = ei;
    const int* dstp = ei + Ee;

    // workspace layout (floats); regions reused across the serialized stream
    float* ws    = (float*)d_ws;
    float* h1    = ws;                 // Nn*128 : layer1 h  (reused as layer2 h)
    float* feat2 = h1 + Nn * 128;      // Nn*128 : layer1 agg/in2 (reused as layer2 agg)
    float* aS    = feat2 + Nn * 128;   // Nn*4
    float* aD    = aS + Nn * 4;        // Nn*4
    float* mM    = aD + Nn * 4;        // Nn*4
    float* dn    = mM + Nn * 4;        // Nn*4
    float* ev    = dn + Nn * 4;        // TE*4  (reused TE*1 for layer2)
    float* Wt1   = ev + TE * 4;        // 128*128 : W1 transposed
    float* Wt2   = Wt1 + 128 * 128;    // 128*64  : W2 transposed

    const float NEG_INF = -__builtin_huge_valf();

    // weight transposes (tiny; makes both GEMM operand streams contiguous)
    k_transpose<<<blocks_for(128 * 128, 256), 256, 0, stream>>>(W1, Wt1, 128, 128);
    k_transpose<<<blocks_for(128 * 64, 256), 256, 0, stream>>>(W2, Wt2, 128, 64);

    // ---------------- layer 1 (H=4, C=32, D=128) ----------------
    {
        dim3 g((unsigned)((Nn + 15) / 16), 2);
        k_gemm_wmma<128><<<g, 128, 0, stream>>>(x, Wt1, h1, (int)Nn);
    }
    k_alpha<<<blocks_for(Nn * 4, 256), 256, 0, stream>>>(h1, as1, ad1, aS, aD, Nn, 4, 32);
    k_fill<<<blocks_for(Nn * 4, 256), 256, 0, stream>>>(mM, NEG_INF, Nn * 4);
    k_fill<<<blocks_for(Nn * 4, 256), 256, 0, stream>>>(dn, 0.0f, Nn * 4);
    k_fill<<<blocks_for(Nn * 128, 256), 256, 0, stream>>>(feat2, 0.0f, Nn * 128);
    k_edge_max<<<blocks_for(TE * 4, 256), 256, 0, stream>>>(srcp, dstp, Ee, Nn, 4, aS, aD, ev, mM);
    k_edge_exp<<<blocks_for(TE * 4, 256), 256, 0, stream>>>(srcp, dstp, Ee, Nn, 4, mM, ev, dn);
    k_edge_scatter<<<(unsigned)TE, 128, 0, stream>>>(srcp, dstp, Ee, Nn, 4, 128, h1, ev, dn, feat2);
    k_bias_relu<<<blocks_for(Nn * 128, 256), 256, 0, stream>>>(feat2, b1, Nn, 128);

    // ---------------- layer 2 (H=1, C=64, D=64) ----------------
    {
        dim3 g((unsigned)((Nn + 15) / 16), 1);
        k_gemm_wmma<64><<<g, 128, 0, stream>>>(feat2, Wt2, h1, (int)Nn);
    }
    k_alpha<<<blocks_for(Nn, 256), 256, 0, stream>>>(h1, as2, ad2, aS, aD, Nn, 1, 64);
    k_fill<<<blocks_for(Nn, 256), 256, 0, stream>>>(mM, NEG_INF, Nn);
    k_fill<<<blocks_for(Nn, 256), 256, 0, stream>>>(dn, 0.0f, Nn);
    k_fill<<<blocks_for(Nn * 64, 256), 256, 0, stream>>>(feat2, 0.0f, Nn * 64);
    k_edge_max<<<blocks_for(TE, 256), 256, 0, stream>>>(srcp, dstp, Ee, Nn, 1, aS, aD, ev, mM);
    k_edge_exp<<<blocks_for(TE, 256), 256, 0, stream>>>(srcp, dstp, Ee, Nn, 1, mM, ev, dn);
    k_edge_scatter<<<(unsigned)TE, 64, 0, stream>>>(srcp, dstp, Ee, Nn, 1, 64, h1, ev, dn, feat2);

    // bias + relu + L2 normalize -> d_out
    k_final<<<blocks_for(Nn * 32, 128), 128, 0, stream>>>(feat2, b2, out, Nn);
}